// YOLO1_13331578486946
// MI455X (gfx1250) — compile-verified
//
#include <hip/hip_runtime.h>
#include <stdint.h>

// Problem geometry (fixed by the reference).
#define BATCH 64
#define CCH   31
#define HF    28
#define WF    28
#define NCELL 784          // 28*28
#define NCLS  21
#define BG    20
#define NW    25           // ceil(784/32) mask words per row
#define CLSOFF 4096.0f
#define NMS_THR 0.5f

typedef unsigned int v4u  __attribute__((ext_vector_type(4)));
typedef int          v8i_t __attribute__((ext_vector_type(8)));
typedef int          v4i_t __attribute__((ext_vector_type(4)));

__device__ __forceinline__ float sigmoidf_(float x) {
    return 1.0f / (1.0f + expf(-x));
}

__global__ void __launch_bounds__(512)
yolo_post_kernel(const float* __restrict__ feat,
                 const int* __restrict__ hptr,
                 const int* __restrict__ wptr,
                 float* __restrict__ out)
{
    // 97,216 B feature stage; aliased by the 78,400 B NMS bitmask matrix later.
    __shared__ float lds_feat[CCH * NCELL];
    __shared__ float bx1[NCELL], by1[NCELL], bx2[NCELL], by2[NCELL];
    __shared__ float catoff[NCELL];
    __shared__ float svals[NCELL];
    __shared__ int   order[NCELL];
    __shared__ float sx1[NCELL], sy1[NCELL], sx2[NCELL], sy2[NCELL], sarea[NCELL];
    __shared__ unsigned char validb[NCELL], valid_s[NCELL], keepf[NCELL];

    unsigned* masks = (unsigned*)lds_feat;   // reuse after decode phase

    const int b   = blockIdx.x;
    const int tid = threadIdx.x;
    const int nth = blockDim.x;
    const float* gsrc = feat + (size_t)b * (CCH * NCELL);

    // ---------------- Phase 0: TDM-stage features[b] into LDS ----------------
#if __has_builtin(__builtin_amdgcn_tensor_load_to_lds) && __has_builtin(__builtin_amdgcn_s_wait_tensorcnt)
    if (tid < 32) {   // one wave issues the DMA once
        uint64_t ga = (uint64_t)(uintptr_t)gsrc;
        uint32_t la = (uint32_t)(uintptr_t)(&lds_feat[0]);   // low 32 bits = LDS offset
        v4u g0;
        g0.x = 1u;                                           // count=1 (valid user D#)
        g0.y = la;                                           // lds_addr
        g0.z = (uint32_t)(ga & 0xffffffffu);                 // global_addr[31:0]
        g0.w = (uint32_t)((ga >> 32) & 0x01ffffffu)          // global_addr[56:32]
             | (2u << 30);                                   // type=2 ("image")
        v8i_t g1;
        g1[0] = (int)(2u << 16);                             // data_size=4B, mask=0
        g1[1] = (int)((uint32_t)(NCELL & 0xffff) << 16);     // tensor_dim0 lo16
        g1[2] = (int)(((uint32_t)NCELL >> 16)                // tensor_dim0 hi16
                    | ((uint32_t)(CCH & 0xffff) << 16));     // tensor_dim1 lo16
        g1[3] = (int)(((uint32_t)CCH >> 16)                  // tensor_dim1 hi16
                    | ((uint32_t)NCELL << 16));              // tile_dim0 = 784
        g1[4] = (int)((uint32_t)CCH);                        // tile_dim1=31, tile_dim2=0
        g1[5] = (int)((uint32_t)NCELL);                      // tensor_dim0_stride lo32 = 784
        g1[6] = 0;                                           // stride hi / dim1_stride
        g1[7] = 0;
        v4i_t g2 = {0, 0, 0, 0}, g3 = {0, 0, 0, 0};
        v8i_t g4 = {0, 0, 0, 0, 0, 0, 0, 0};                 // unused trailing group (6-arg form)
        __builtin_amdgcn_tensor_load_to_lds(g0, g1, g2, g3, g4, 0);
        __builtin_amdgcn_s_wait_tensorcnt(0);                // data resident in LDS
    }
#else
    for (int i = tid; i < CCH * NCELL; i += nth) lds_feat[i] = gsrc[i];
#endif
    __syncthreads();

    // ---------------- Phase 1: per-cell decode -------------------------------
    const float wori = (float)wptr[0];
    const float hori = (float)hptr[0];
    const float wsc  = wori / (float)WF;
    const float hsc  = hori / (float)HF;

    for (int n = tid; n < NCELL; n += nth) {
        float v[CCH];
        #pragma unroll
        for (int c = 0; c < CCH; ++c) v[c] = lds_feat[c * NCELL + n];

        // confidence max / argmax over the two anchors (first-max tie break)
        float c0 = v[4], c1 = v[9];
        float cmax = fmaxf(c0, c1);
        int   carg = (c1 > c0) ? 1 : 0;

        int sel = carg * 5;
        float bxc = sigmoidf_(v[sel + 0]);
        float byc = sigmoidf_(v[sel + 1]);
        float bwv = sigmoidf_(v[sel + 2]);
        float bhv = sigmoidf_(v[sel + 3]);

        // softmax over 21 classes, then prob/cat of cmax * softmax
        float m = v[10];
        #pragma unroll
        for (int c = 11; c < CCH; ++c) m = fmaxf(m, v[c]);
        float e[NCLS]; float s = 0.0f;
        #pragma unroll
        for (int c = 0; c < NCLS; ++c) { e[c] = expf(v[10 + c] - m); s += e[c]; }
        float inv = 1.0f / s;
        float best = -INFINITY; int bc = 0;
        #pragma unroll
        for (int c = 0; c < NCLS; ++c) {
            float cp = cmax * (e[c] * inv);
            if (cp > best) { best = cp; bc = c; }   // strict > == first occurrence
        }

        float col = (float)(n % WF), row = (float)(n / WF);
        float cx = (bxc + col) * wsc;
        float cy = (byc + row) * hsc;
        float w2 = bwv * wori * 0.5f;
        float h2 = bhv * hori * 0.5f;
        float x1 = cx - w2, y1 = cy - h2, x2 = cx + w2, y2 = cy + h2;

        bx1[n] = x1; by1[n] = y1; bx2[n] = x2; by2[n] = y2;
        catoff[n] = (float)bc * CLSOFF;
        int valid = (bc != BG);
        validb[n] = (unsigned char)valid;
        svals[n]  = valid ? best : -INFINITY;

        size_t ob = ((size_t)b * NCELL + n) * 7;
        out[ob + 0] = x1; out[ob + 1] = y1; out[ob + 2] = x2; out[ob + 3] = y2;
        out[ob + 4] = best; out[ob + 5] = (float)bc;
    }
    __syncthreads();

    // ---------------- Phase 2: stable descending rank sort -------------------
    for (int n = tid; n < NCELL; n += nth) {
        float si = svals[n];
        int r = 0;
        for (int j = 0; j < NCELL; ++j) {
            float sj = svals[j];
            r += ((sj > si) || ((sj == si) && (j < n))) ? 1 : 0;
        }
        order[r] = n;      // ranks unique -> race-free permutation
    }
    __syncthreads();

    // ---------------- Phase 3: gather sorted class-offset boxes --------------
    for (int i = tid; i < NCELL; i += nth) {
        int n = order[i];
        float off = catoff[n];
        float x1 = bx1[n] + off, y1 = by1[n] + off;
        float x2 = bx2[n] + off, y2 = by2[n] + off;
        sx1[i] = x1; sy1[i] = y1; sx2[i] = x2; sy2[i] = y2;
        sarea[i] = fmaxf(x2 - x1, 0.0f) * fmaxf(y2 - y1, 0.0f);
        valid_s[i] = validb[n];
    }
    __syncthreads();   // lds_feat reads all done before masks overwrite it

    // ---------------- Phase 4: IoU suppression bitmask (784 x 25 words) ------
    for (int task = tid; task < NCELL * NW; task += nth) {
        int i = task / NW, w = task % NW;
        float x1 = sx1[i], y1 = sy1[i], x2 = sx2[i], y2 = sy2[i], ai = sarea[i];
        unsigned bits = 0u;
        int jb = w * 32;
        for (int k = 0; k < 32; ++k) {
            int j = jb + k;
            if (j >= NCELL) break;
            if (j <= i) continue;
            float xx1 = fmaxf(x1, sx1[j]);
            float yy1 = fmaxf(y1, sy1[j]);
            float xx2 = fminf(x2, sx2[j]);
            float yy2 = fminf(y2, sy2[j]);
            float inter = fmaxf(xx2 - xx1, 0.0f) * fmaxf(yy2 - yy1, 0.0f);
            float uni = ai + sarea[j] - inter;
            float iou = inter / fmaxf(uni, 1e-9f);
            if (iou > NMS_THR) bits |= (1u << k);
        }
        masks[i * NW + w] = bits;
    }
    __syncthreads();

    // ---------------- Phase 5: single-wave greedy suppression ----------------
    if (tid < 32) {
        int lane = tid;
        unsigned removed = 0u;        // lane l owns removed-word l (l < 25)
        for (int i = 0; i < NCELL; ++i) {
            unsigned rw = (unsigned)__shfl((int)removed, i >> 5, 32);
            int kept = (valid_s[i] != 0) && !((rw >> (i & 31)) & 1u);
            if (kept && lane < NW) removed |= masks[i * NW + lane];
            if (lane == 0) keepf[i] = (unsigned char)kept;
        }
    }
    __syncthreads();

    // ---------------- Phase 6: scatter keep flags ----------------------------
    for (int i = tid; i < NCELL; i += nth) {
        int n = order[i];
        out[((size_t)b * NCELL + n) * 7 + 6] = keepf[i] ? 1.0f : 0.0f;
    }
}

extern "C" void kernel_launch(void* const* d_in, const int* in_sizes, int n_in,
                              void* d_out, int out_size, void* d_ws, size_t ws_size,
                              hipStream_t stream) {
    (void)in_sizes; (void)n_in; (void)out_size; (void)d_ws; (void)ws_size;
    const float* feat = (const float*)d_in[0];
    const int*   hptr = (const int*)d_in[1];
    const int*   wptr = (const int*)d_in[2];
    float*       out  = (float*)d_out;
    yolo_post_kernel<<<BATCH, 512, 0, stream>>>(feat, hptr, wptr, out);
}